// SLModel_89472758710968
// MI455X (gfx1250) — compile-verified
//
#include <hip/hip_runtime.h>

// CRF NLL with batch-coupled logsumexp quirks, tuned for MI455X (gfx1250).
// 32 persistent workgroups x 256 threads; state in LDS; per-step device-wide
// reduction via L2 atomics + per-step counters; exp-space GEMM on
// v_wmma_f32_16x16x4_f32 (wave-specialized against the max-plus VALU work).

#define T_     1024
#define L_     32
#define START_ 30
#define NEG_   (-10000.0f)
#define NWG    32
#define NTH    256
#define ROWS   32   // batch rows per workgroup (NWG*ROWS == B == 1024)

typedef __attribute__((ext_vector_type(2))) float v2f;
typedef __attribute__((ext_vector_type(8))) float v8f;

__global__ void __launch_bounds__(256) ws_zero_kernel(unsigned* p, int n) {
  int i = blockIdx.x * blockDim.x + threadIdx.x;
  if (i < n) p[i] = 0u;
}

__global__ void __launch_bounds__(NTH) crf_fwd_kernel(
    const float* __restrict__ emission,   // (B, T, L) f32
    const int*   __restrict__ labels,     // (B, T) i32
    const float* __restrict__ mask,       // (B, T) f32
    const float* __restrict__ trans,      // (L, L) f32
    float*       __restrict__ out,        // (B,) f32
    float*       __restrict__ s_acc,      // (T, L) per-step global accumulators
    unsigned*    __restrict__ ctr,        // (T,) per-step arrival counters
    float*       __restrict__ S_final,    // scalar
    unsigned*    __restrict__ ctr_final)  // scalar
{
  __shared__ float trans_s[L_][L_ + 1];
  __shared__ float expT_s [L_][L_ + 1];
  __shared__ float fwd_s  [ROWS][L_ + 1];
  __shared__ float a_s    [ROWS][L_ + 1];
  __shared__ float mp_s   [ROWS][L_ + 1];
  __shared__ float P_s    [ROWS][L_ + 1];
  __shared__ __align__(16) float emit_s[ROWS][L_];
  __shared__ float u_s[ROWS], mk_s[ROWS], path_s[ROWS];
  __shared__ float C_s[L_], spart_s[L_];
  __shared__ int   prev_s[ROWS];

  const int tid = threadIdx.x;
  const int b0  = blockIdx.x * ROWS;

  // ---- one-time setup: transitions (raw + exp), fwd init, path t=0 ----
  #pragma unroll
  for (int e = tid; e < L_ * L_; e += NTH) {
    int i = e >> 5, j = e & 31;
    float v = trans[i * L_ + j];
    trans_s[i][j] = v;
    expT_s[i][j]  = __expf(v);           // exp(-10000) flushes to 0: fine
  }
  #pragma unroll
  for (int e = tid; e < ROWS * L_; e += NTH) {
    int b = e >> 5, i = e & 31;
    fwd_s[b][i] = (i == START_) ? 0.0f : NEG_;
  }
  if (tid < ROWS) {
    int gb = b0 + tid;
    int l0 = labels[(size_t)gb * T_];
    float mk0 = mask[(size_t)gb * T_];
    path_s[tid] = mk0 * emission[(size_t)gb * T_ * L_ + l0];
    prev_s[tid] = l0;
  }
  __syncthreads();

  for (int t = 1; t < T_; ++t) {
    // A: stream emission tile (one float4 per thread) + row stats
    {
      int b = tid >> 3, q = (tid & 7) << 2;
      const float4 v =
          *(const float4*)(emission + ((size_t)(b0 + b) * T_ + t) * L_ + q);
      *(float4*)&emit_s[b][q] = v;
      if (t + 1 < T_)
        __builtin_prefetch(emission + ((size_t)(b0 + b) * T_ + t + 1) * L_ + q, 0, 0);
    }
    if (tid < ROWS) {
      float mx = fwd_s[tid][0];
      #pragma unroll
      for (int j = 1; j < L_; ++j) mx = fmaxf(mx, fwd_s[tid][j]);
      u_s[tid]   = mx;
      mk_s[tid]  = mask[(size_t)(b0 + tid) * T_ + t];
      spart_s[tid] = 0.0f;
    }
    __syncthreads();

    // B: A-matrix in exp space (stable: fwd - rowmax <= 0)
    #pragma unroll
    for (int e = tid; e < ROWS * L_; e += NTH) {
      int b = e >> 5, j = e & 31;
      a_s[b][j] = __expf(fwd_s[b][j] - u_s[b]);
    }
    __syncthreads();

    // C: wave specialization. Waves 0-3: P = A(32x32) @ expT^T via f32 WMMA
    //    (one 16x16 output tile per wave, K=32 as 8 chunks of 4).
    //    Waves 4-7: max-plus mp[b,i] = max_j(fwd[b,j] + mk*T[i,j]).
    const int wave = tid >> 5;
    if (wave < 4) {
      const int btile = wave >> 1, itile = wave & 1;
      const int lane  = tid & 31;
      const int Mrow  = lane & 15;
      const int khalf = (lane >> 4) << 1;          // lanes 16-31 hold K+2,K+3
      v8f c = {};
      #pragma unroll
      for (int kc = 0; kc < 8; ++kc) {
        const int kb = (kc << 2) + khalf;
        v2f af, bf;
        af[0] = a_s[btile * 16 + Mrow][kb];
        af[1] = a_s[btile * 16 + Mrow][kb + 1];
        bf[0] = expT_s[itile * 16 + Mrow][kb];     // B[K][N] = expT[N][K]
        bf[1] = expT_s[itile * 16 + Mrow][kb + 1];
        c = __builtin_amdgcn_wmma_f32_16x16x4_f32(false, af, false, bf,
                                                  (short)0, c, false, false);
      }
      const int N  = lane & 15;
      const int Mb = (lane >> 4) << 3;             // lanes 16-31 hold M=8..15
      #pragma unroll
      for (int r = 0; r < 8; ++r)
        P_s[btile * 16 + Mb + r][itile * 16 + N] = c[r];
    } else {
      #pragma unroll
      for (int e = tid - 128; e < ROWS * L_; e += 128) {
        int b = e >> 5, i = e & 31;
        float mk = mk_s[b];
        float mx = fmaf(mk, trans_s[i][0], fwd_s[b][0]);
        #pragma unroll
        for (int j = 1; j < L_; ++j)
          mx = fmaxf(mx, fmaf(mk, trans_s[i][j], fwd_s[b][j]));
        mp_s[b][i] = mx;
      }
    }
    __syncthreads();

    // D: per-label partial sums over this WG's 32 batch rows (ds_add_f32)
    #pragma unroll
    for (int e = tid; e < ROWS * L_; e += NTH) {
      int b = e >> 5, i = e & 31;
      float contrib = __expf(u_s[b] - mp_s[b][i]) * P_s[b][i];
      atomicAdd(&spart_s[i], contrib);
    }
    __syncthreads();

    // E: device-wide reduce + sync; path-score update overlapped on wave 1
    if (tid < L_)
      atomicAdd(&s_acc[(size_t)t * L_ + tid], spart_s[tid]);
    if (tid >= 32 && tid < 32 + ROWS) {
      int b = tid - 32, gb = b0 + b;
      int l = labels[(size_t)gb * T_ + t];
      float mk  = mk_s[b];
      float mkp = mask[(size_t)gb * T_ + t - 1];
      path_s[b] += mk * emit_s[b][l] + mkp * trans_s[l][prev_s[b]];
      prev_s[b] = l;
    }
    if (tid == 0) {
      __threadfence();
      __hip_atomic_fetch_add(&ctr[t], 1u, __ATOMIC_ACQ_REL,
                             __HIP_MEMORY_SCOPE_AGENT);
      while (__hip_atomic_load(&ctr[t], __ATOMIC_ACQUIRE,
                               __HIP_MEMORY_SCOPE_AGENT) < NWG)
        __builtin_amdgcn_s_sleep(1);
    }
    if (tid < L_) {
      float S = __hip_atomic_load(&s_acc[(size_t)t * L_ + tid],
                                  __ATOMIC_ACQUIRE, __HIP_MEMORY_SCOPE_AGENT);
      C_s[tid] = __logf(S);
    }
    __syncthreads();

    // F: fwd' = mk*emit + mp + C (broadcast C couples the batch)
    #pragma unroll
    for (int e = tid; e < ROWS * L_; e += NTH) {
      int b = e >> 5, i = e & 31;
      fwd_s[b][i] = fmaf(mk_s[b], emit_s[b][i], mp_s[b][i] + C_s[i]);
    }
    __syncthreads();
  }

  // ---- final quirked LSE: partition[b] = rowmax[b] + log(sum_all exp(fwd-rowmax)) ----
  if (tid < ROWS) {
    float mx = fwd_s[tid][0];
    #pragma unroll
    for (int j = 1; j < L_; ++j) mx = fmaxf(mx, fwd_s[tid][j]);
    u_s[tid] = mx;
    if (tid == 0) spart_s[0] = 0.0f;
  }
  __syncthreads();
  #pragma unroll
  for (int e = tid; e < ROWS * L_; e += NTH) {
    int b = e >> 5, i = e & 31;
    atomicAdd(&spart_s[0], __expf(fwd_s[b][i] - u_s[b]));
  }
  __syncthreads();
  if (tid == 0) {
    atomicAdd(S_final, spart_s[0]);
    __threadfence();
    __hip_atomic_fetch_add(ctr_final, 1u, __ATOMIC_ACQ_REL,
                           __HIP_MEMORY_SCOPE_AGENT);
    while (__hip_atomic_load(ctr_final, __ATOMIC_ACQUIRE,
                             __HIP_MEMORY_SCOPE_AGENT) < NWG)
      __builtin_amdgcn_s_sleep(1);
  }
  __syncthreads();
  if (tid < ROWS) {
    float S = __hip_atomic_load(S_final, __ATOMIC_ACQUIRE,
                                __HIP_MEMORY_SCOPE_AGENT);
    out[b0 + tid] = (u_s[tid] + __logf(S)) - path_s[tid];  // -(path - partition)
  }
}

extern "C" void kernel_launch(void* const* d_in, const int* in_sizes, int n_in,
                              void* d_out, int out_size, void* d_ws, size_t ws_size,
                              hipStream_t stream) {
  (void)in_sizes; (void)n_in; (void)out_size; (void)ws_size;
  const float* emission = (const float*)d_in[0];
  const int*   labels   = (const int*)d_in[1];
  const float* mask     = (const float*)d_in[2];
  const float* trans    = (const float*)d_in[3];
  float* out = (float*)d_out;

  float*    s_acc     = (float*)d_ws;                                  // T*L f32
  unsigned* ctr       = (unsigned*)((char*)d_ws + (size_t)T_ * L_ * 4); // T u32
  float*    S_final   = (float*)((char*)ctr + (size_t)T_ * 4);          // 1 f32
  unsigned* ctr_final = (unsigned*)((char*)S_final + 4);                // 1 u32

  const int nzero = T_ * L_ + T_ + 2;   // everything above, in dwords
  ws_zero_kernel<<<(nzero + 255) / 256, 256, 0, stream>>>((unsigned*)d_ws, nzero);
  crf_fwd_kernel<<<NWG, NTH, 0, stream>>>(emission, labels, mask, trans, out,
                                          s_acc, ctr, S_final, ctr_final);
}